// Classifier_74019466379909
// MI455X (gfx1250) — compile-verified
//
#include <hip/hip_runtime.h>

typedef __attribute__((ext_vector_type(2))) float v2f;
typedef __attribute__((ext_vector_type(8))) float v8f;

#define N_NODES 100000
#define N_EDGES 1600000
#define N_GRAPHS 1000
#define IN_DIM 95
#define K1PAD 96
#define HID 128
#define N_CLS 16

// ---------------------------------------------------------------------------
// Degree accumulation: one thread per edge, fp32 atomic adds (L2 atomics).
// ---------------------------------------------------------------------------
__global__ void deg_kernel(const int* __restrict__ src, const int* __restrict__ dst,
                           float* __restrict__ deg_src, float* __restrict__ deg_dst, int E) {
    int e = blockIdx.x * blockDim.x + threadIdx.x;
    if (e < E) {
        atomicAdd(&deg_src[src[e]], 1.0f);
        atomicAdd(&deg_dst[dst[e]], 1.0f);
    }
}

// deg -> deg^{-1/2} with clip at 1 (in place; norm_src and norm_dst are
// contiguous, one pass covers both).
__global__ void norm_kernel(float* __restrict__ d, int n) {
    int i = blockIdx.x * blockDim.x + threadIdx.x;
    if (i < n) d[i] = rsqrtf(fmaxf(d[i], 1.0f));
}

// Zero-padded copy of W1 (95x128) into Wp (96x128).
__global__ void padW1_kernel(const float* __restrict__ W1, float* __restrict__ Wp) {
    int i = blockIdx.x * blockDim.x + threadIdx.x;   // [0, 96*128)
    int k = i >> 7;
    Wp[i] = (k < IN_DIM) ? W1[i] : 0.0f;
}

// ---------------------------------------------------------------------------
// Edge scatter: agg[dst] += h[src] * norm_src[src].
// Two edges per 256-thread block; one lane per feature column. h and agg
// (51 MB) are L2-resident, so the atomics resolve in L2.
// ---------------------------------------------------------------------------
__global__ void scatter_kernel(const float* __restrict__ h,
                               const float* __restrict__ norm_src,
                               const int* __restrict__ src, const int* __restrict__ dst,
                               float* __restrict__ agg, int F, int ldh, int ldagg, int E) {
    int e = blockIdx.x * 2 + (threadIdx.x >> 7);
    int f = threadIdx.x & 127;
    if (e >= E || f >= F) return;
    int s = src[e];
    int d = dst[e];
    atomicAdd(&agg[(size_t)d * ldagg + f], h[(size_t)s * ldh + f] * norm_src[s]);
}

// ---------------------------------------------------------------------------
// Fused GEMM: C = relu( (rowscale[:,None] * A) @ B + bias ) with
// V_WMMA_F32_16X16X4_F32 (exact fp32 path). K is a compile-time multiple of 4
// and A/B are pre-padded, so the inner loop is branch-free straight-line
// code: one global_load_b64 (A pair) + two global_load_b32 (B rows) per WMMA.
//
// One wave32 owns a 16x16 output tile. Fragment layouts per CDNA5 ISA 7.12.2:
//   A 16x4 fp32 : lanes 0-15 -> M=lane, {K0,K1}; lanes 16-31 -> M=lane-16, {K2,K3}
//   B 4x16 fp32 : mirrored; N = lane&15
//   C/D 16x16   : VGPR v -> M = v + 8*(lane>=16), N = lane&15
// Grid covers tiles exactly (M=100000=6250*16, N=128=8*16) so EXEC is
// all-ones throughout, as WMMA requires.
// ---------------------------------------------------------------------------
template <int K>
__global__ void __launch_bounds__(256)
gemm_wmma_kernel(const float* __restrict__ A,        // M x K, stride K
                 const float* __restrict__ B,        // K x HID, stride HID
                 const float* __restrict__ bias,     // HID
                 const float* __restrict__ rowscale, // M
                 float* __restrict__ C,              // M x HID, stride HID
                 int do_relu) {
    int wave = threadIdx.x >> 5;
    int lane = threadIdx.x & 31;
    int tile = blockIdx.x * 8 + wave;   // 8 waves per block
    int mt = tile >> 3;                 // tilesN = HID/16 = 8
    int nt = tile & 7;
    int m0 = mt << 4;
    int n0 = nt << 4;

    int half = lane >> 4;   // 0: K pair (k,k+1); 1: K pair (k+2,k+3)
    int l    = lane & 15;   // M index for A, N index for B/C/D

    int row = m0 + l;
    float rs = rowscale[row];
    const float* __restrict__ Arow = A + (size_t)row * K + 2 * half;
    const float* __restrict__ Bcol = B + (size_t)(2 * half) * HID + n0 + l;

    v8f acc = {};
#pragma unroll 8
    for (int k = 0; k < K; k += 4) {
        v2f a, b;
        const v2f av = *(const v2f*)(Arow + k);      // global_load_b64
        a.x = av.x * rs;
        a.y = av.y * rs;
        b.x = Bcol[(size_t)k * HID];                 // row k+2*half
        b.y = Bcol[(size_t)(k + 1) * HID];           // row k+2*half+1
        // 8 args: (neg_a, A, neg_b, B, c_mod, C, reuse_a, reuse_b)
        acc = __builtin_amdgcn_wmma_f32_16x16x4_f32(
            false, a, false, b, (short)0, acc, false, false);
    }

    float bn = bias[n0 + l];
#pragma unroll
    for (int v = 0; v < 8; ++v) {
        int mm = m0 + v + 8 * half;
        float val = acc[v] + bn;
        if (do_relu) val = fmaxf(val, 0.0f);
        C[(size_t)mm * HID + n0 + l] = val;
    }
}

// ---------------------------------------------------------------------------
// Graph mean-pool part 1: atomic segment sums + counts.
// Grid covers exactly N_NODES*HID elements (divisible by 256).
// ---------------------------------------------------------------------------
__global__ void pool_kernel(const float* __restrict__ h, const int* __restrict__ gid,
                            float* __restrict__ gsums, float* __restrict__ gcnt) {
    int i = blockIdx.x * blockDim.x + threadIdx.x;   // [0, N_NODES*HID)
    int n = i >> 7;
    int f = i & 127;
    int g = gid[n];
    atomicAdd(&gsums[(size_t)g * HID + f], h[i]);
    if (f == 0) atomicAdd(&gcnt[g], 1.0f);
}

// ---------------------------------------------------------------------------
// Classifier: out[g,c] = (gsums[g,:]/cnt[g]) @ Wc[:,c] + bc[c]. 2 MFLOP total.
// ---------------------------------------------------------------------------
__global__ void cls_kernel(const float* __restrict__ gsums, const float* __restrict__ gcnt,
                           const float* __restrict__ Wc, const float* __restrict__ bc,
                           float* __restrict__ out) {
    int t = blockIdx.x * blockDim.x + threadIdx.x;
    if (t >= N_GRAPHS * N_CLS) return;
    int g = t >> 4;
    int c = t & 15;
    float inv = 1.0f / fmaxf(gcnt[g], 1.0f);
    float acc = 0.0f;
#pragma unroll 8
    for (int k = 0; k < HID; ++k)
        acc = fmaf(gsums[(size_t)g * HID + k], Wc[k * N_CLS + c], acc);
    out[t] = acc * inv + bc[c];
}

extern "C" void kernel_launch(void* const* d_in, const int* in_sizes, int n_in,
                              void* d_out, int out_size, void* d_ws, size_t ws_size,
                              hipStream_t stream) {
    const float* x   = (const float*)d_in[0];
    const int*   src = (const int*)  d_in[1];
    const int*   dst = (const int*)  d_in[2];
    const int*   gid = (const int*)  d_in[3];
    const float* W1  = (const float*)d_in[4];
    const float* b1  = (const float*)d_in[5];
    const float* W2  = (const float*)d_in[6];
    const float* b2  = (const float*)d_in[7];
    const float* W3  = (const float*)d_in[8];
    const float* b3  = (const float*)d_in[9];
    const float* Wc  = (const float*)d_in[10];
    const float* bc  = (const float*)d_in[11];
    float* out = (float*)d_out;

    float* ws       = (float*)d_ws;
    float* norm_src = ws;                               // N_NODES
    float* norm_dst = ws + N_NODES;                     // N_NODES (contiguous)
    float* buf0     = ws + 2 * (size_t)N_NODES;         // N_NODES*HID (h)
    float* buf1     = buf0 + (size_t)N_NODES * HID;     // N_NODES*HID (agg, >= N_NODES*K1PAD)
    float* gsums    = buf1 + (size_t)N_NODES * HID;     // N_GRAPHS*HID
    float* gcnt     = gsums + (size_t)N_GRAPHS * HID;   // N_GRAPHS (contiguous)
    float* W1p      = gcnt + N_GRAPHS;                  // K1PAD*HID

    const int GEMM_BLOCKS    = (N_NODES / 16) * (HID / 16) / 8;  // 6250
    const int SCATTER_BLOCKS = (N_EDGES + 1) / 2;                // 2 edges per block

    // degrees -> norms; pad W1 while that runs
    hipMemsetAsync(norm_src, 0, 2 * (size_t)N_NODES * sizeof(float), stream);
    deg_kernel<<<(N_EDGES + 255) / 256, 256, 0, stream>>>(src, dst, norm_src, norm_dst, N_EDGES);
    norm_kernel<<<(2 * N_NODES + 255) / 256, 256, 0, stream>>>(norm_src, 2 * N_NODES);
    padW1_kernel<<<(K1PAD * HID) / 256, 256, 0, stream>>>(W1, W1p);

    // layer 1: x (100000x95) -> buf0 (100000x128); agg padded to stride 96
    hipMemsetAsync(buf1, 0, (size_t)N_NODES * K1PAD * sizeof(float), stream);
    scatter_kernel<<<SCATTER_BLOCKS, 256, 0, stream>>>(x, norm_src, src, dst, buf1,
                                                       IN_DIM, IN_DIM, K1PAD, N_EDGES);
    gemm_wmma_kernel<K1PAD><<<GEMM_BLOCKS, 256, 0, stream>>>(buf1, W1p, b1, norm_dst, buf0, 1);

    // layer 2
    hipMemsetAsync(buf1, 0, (size_t)N_NODES * HID * sizeof(float), stream);
    scatter_kernel<<<SCATTER_BLOCKS, 256, 0, stream>>>(buf0, norm_src, src, dst, buf1,
                                                       HID, HID, HID, N_EDGES);
    gemm_wmma_kernel<HID><<<GEMM_BLOCKS, 256, 0, stream>>>(buf1, W2, b2, norm_dst, buf0, 1);

    // layer 3
    hipMemsetAsync(buf1, 0, (size_t)N_NODES * HID * sizeof(float), stream);
    scatter_kernel<<<SCATTER_BLOCKS, 256, 0, stream>>>(buf0, norm_src, src, dst, buf1,
                                                       HID, HID, HID, N_EDGES);
    gemm_wmma_kernel<HID><<<GEMM_BLOCKS, 256, 0, stream>>>(buf1, W3, b3, norm_dst, buf0, 1);

    // mean pool + classifier
    hipMemsetAsync(gsums, 0, ((size_t)N_GRAPHS * HID + N_GRAPHS) * sizeof(float), stream);
    pool_kernel<<<(N_NODES * HID) / 256, 256, 0, stream>>>(buf0, gid, gsums, gcnt);
    cls_kernel<<<(N_GRAPHS * N_CLS + 255) / 256, 256, 0, stream>>>(gsums, gcnt, Wc, bc, out);
}